// AttentionDecoder_26448408609464
// MI455X (gfx1250) — compile-verified
//
#include <hip/hip_runtime.h>
#include <hip/hip_bf16.h>
#include <math.h>

// ---------------- problem constants ----------------
constexpr int kS1 = 8, kS2 = 32, kB = 256, kH = 1024, kE = 512;
constexpr int kIn = 2, kOut = 2, kNL = 4;
constexpr int kP   = kS1 * kS2;     // 256 attention positions
constexpr int kKx0 = kH + kE;       // 1536 (layer-0 input width)
constexpr int k4H  = 4 * kH;        // 4096

typedef __attribute__((ext_vector_type(2))) float v2f;
typedef __attribute__((ext_vector_type(8))) float v8f;

__device__ __forceinline__ float wave_sum(float v) {
  #pragma unroll
  for (int off = 16; off > 0; off >>= 1) v += __shfl_xor(v, off, 32);
  return v;
}
__device__ __forceinline__ float wave_max(float v) {
  #pragma unroll
  for (int off = 16; off > 0; off >>= 1) v = fmaxf(v, __shfl_xor(v, off, 32));
  return v;
}
__device__ __forceinline__ float sigm(float x) { return 1.0f / (1.0f + __expf(-x)); }

// ---------------- stage 1: embedding (relu(input @ W_emb.T + b_emb)) -------
__global__ __launch_bounds__(256) void embed_kernel(
    const float* __restrict__ input, const float* __restrict__ W_emb,
    const float* __restrict__ b_emb, float* __restrict__ xbuf) {
  int idx = blockIdx.x * blockDim.x + threadIdx.x;  // kB*kE
  int b = idx >> 9;        // /512
  int e = idx & 511;
  float v = input[b * kIn + 0] * W_emb[e * kIn + 0] +
            input[b * kIn + 1] * W_emb[e * kIn + 1] + b_emb[e];
  xbuf[(size_t)b * kKx0 + kH + e] = fmaxf(v, 0.0f);
}

// ---------------- stage 2: hdot[b] = h0[0][b,:] . w_h ----------------------
__global__ __launch_bounds__(256) void hdot_kernel(
    const float* __restrict__ h0, const float* __restrict__ W_attn,
    float* __restrict__ hdot) {
  int b    = blockIdx.x * 8 + (threadIdx.x >> 5);
  int lane = threadIdx.x & 31;
  const float* hrow = h0 + (size_t)b * kH;
  const float* wh   = W_attn;  // first H entries
  float acc = 0.f;
  #pragma unroll
  for (int t = 0; t < 8; ++t) {
    int h = (t * 32 + lane) * 4;
    float4 a = *(const float4*)(hrow + h);
    float4 w = *(const float4*)(wh + h);
    acc += a.x * w.x + a.y * w.y + a.z * w.z + a.w * w.w;
  }
  acc = wave_sum(acc);
  if (lane == 0) hdot[b] = acc;
}

// ---------------- stage 3: scoresT[b,p] = ao[p,b,:] . w_y + hdot[b] + ba ---
__global__ __launch_bounds__(256) void scores_kernel(
    const float* __restrict__ all_outputs, const float* __restrict__ W_attn,
    const float* __restrict__ b_attn, const float* __restrict__ hdot,
    float* __restrict__ scoresT) {
  int wid  = blockIdx.x * 8 + (threadIdx.x >> 5);  // 0..65535
  int lane = threadIdx.x & 31;
  int p = wid >> 8;
  int b = wid & 255;
  const float* row = all_outputs + ((size_t)p * kB + b) * kH;
  const float* wy  = W_attn + kH;
  float acc = 0.f;
  #pragma unroll
  for (int t = 0; t < 8; ++t) {
    int h = (t * 32 + lane) * 4;
    float4 a = *(const float4*)(row + h);
    float4 w = *(const float4*)(wy + h);
    acc += a.x * w.x + a.y * w.y + a.z * w.z + a.w * w.w;
  }
  acc = wave_sum(acc);
  if (lane == 0) scoresT[(size_t)b * kP + p] = acc + hdot[b] + b_attn[0];
}

// ---------------- stage 4: row softmax over 256 positions ------------------
__global__ __launch_bounds__(256) void softmax_kernel(
    const float* __restrict__ scoresT, float* __restrict__ out_norm) {
  int b    = blockIdx.x * 8 + (threadIdx.x >> 5);
  int lane = threadIdx.x & 31;
  const float* row = scoresT + (size_t)b * kP;
  float v[8];
  float mx = -INFINITY;
  #pragma unroll
  for (int t = 0; t < 8; ++t) { v[t] = row[t * 32 + lane]; mx = fmaxf(mx, v[t]); }
  mx = wave_max(mx);
  float s = 0.f;
  #pragma unroll
  for (int t = 0; t < 8; ++t) { v[t] = __expf(v[t] - mx); s += v[t]; }
  s = wave_sum(s);
  float inv = 1.0f / s;
  float* orow = out_norm + (size_t)b * kP;
  #pragma unroll
  for (int t = 0; t < 8; ++t) orow[t * 32 + lane] = v[t] * inv;
}

// ---------------- stage 5: final_attn = S1 * sum_i norm[b,224+i]*ao[7,i,b,:]
__global__ __launch_bounds__(256) void applied_kernel(
    const float* __restrict__ all_outputs, const float* __restrict__ norm,
    float* __restrict__ xbuf) {
  int b = blockIdx.y;
  int h = blockIdx.x * 256 + threadIdx.x;
  const float* base = all_outputs + ((size_t)((kS1 - 1) * kS2) * kB + b) * kH + h;
  float acc = 0.f;
  #pragma unroll 4
  for (int i = 0; i < kS2; ++i) {
    float w = norm[(size_t)b * kP + (kS1 - 1) * kS2 + i];
    acc += w * base[(size_t)i * kB * kH];
  }
  xbuf[(size_t)b * kKx0 + h] = (float)kS1 * acc;
}

// ---------------- LSTM layer: fused 4-gate WMMA GEMM + cell update ---------
// One wave owns a 32(batch) x 16(H) output strip across all 4 gates:
// 8 f32 WMMA accumulators (two 16x16 M-tiles x 4 gates). Each K-step loads
// 2 A fragments + 4 B fragments (6x b64) and issues 8 v_wmma_f32_16x16x4_f32.
// A fragment (16x4 f32): lane<16 -> {K=k,k+1} of row M=lane; lane>=16 -> {K=k+2,k+3}.
// B fragment (4x16 f32): lane<16 -> {K=k,k+1} of col N=lane; lane>=16 -> {K=k+2,k+3}.
// C/D (16x16 f32): element r of lane L -> row r+8*(L>=16), col L%16.
#define WMMA_F32(Afrag, Bfrag, Cacc) \
  __builtin_amdgcn_wmma_f32_16x16x4_f32(false, (Afrag), false, (Bfrag), (short)0, (Cacc), false, false)

__device__ __forceinline__ void gemm_phase(
    const float* __restrict__ Xr0, const float* __restrict__ Xr1,
    const float* __restrict__ W0, const float* __restrict__ W1,
    const float* __restrict__ W2, const float* __restrict__ W3,
    int K,
    v8f& a00, v8f& a01, v8f& a02, v8f& a03,
    v8f& a10, v8f& a11, v8f& a12, v8f& a13) {
  // warm the streams (global_prefetch_b8); outside the hot loop on purpose
  __builtin_prefetch(Xr0, 0, 3);
  __builtin_prefetch(Xr1, 0, 3);
  __builtin_prefetch(W0, 0, 3);
  __builtin_prefetch(W1, 0, 3);
  __builtin_prefetch(W2, 0, 3);
  __builtin_prefetch(W3, 0, 3);
  for (int k = 0; k < K; k += 4) {
    v2f av0 = *(const v2f*)(Xr0 + k);
    v2f av1 = *(const v2f*)(Xr1 + k);
    v2f b0  = *(const v2f*)(W0 + k);
    v2f b1  = *(const v2f*)(W1 + k);
    v2f b2  = *(const v2f*)(W2 + k);
    v2f b3  = *(const v2f*)(W3 + k);
    a00 = WMMA_F32(av0, b0, a00);
    a10 = WMMA_F32(av1, b0, a10);
    a01 = WMMA_F32(av0, b1, a01);
    a11 = WMMA_F32(av1, b1, a11);
    a02 = WMMA_F32(av0, b2, a02);
    a12 = WMMA_F32(av1, b2, a12);
    a03 = WMMA_F32(av0, b3, a03);
    a13 = WMMA_F32(av1, b3, a13);
  }
}

__global__ __launch_bounds__(256) void lstm_layer_kernel(
    const float* __restrict__ X, int Kx,
    const float* __restrict__ Hprev, const float* __restrict__ Cprev,
    const float* __restrict__ Wih, const float* __restrict__ Whh,
    const float* __restrict__ bih, const float* __restrict__ bhh,
    float* __restrict__ Hout, float* __restrict__ Cout) {
  const int wid  = blockIdx.x * 8 + (threadIdx.x >> 5);  // 512 waves total
  const int lane = threadIdx.x & 31;
  const int hi   = lane >> 4;   // 0/1: which K/M half this lane serves
  const int lp   = lane & 15;
  const int nt   = wid & 63;    // 64 tiles along H
  const int mt   = wid >> 6;    // 8 strips of 32 batch rows
  const int n0   = nt * 16;

  v8f a00 = {}, a01 = {}, a02 = {}, a03 = {};
  v8f a10 = {}, a11 = {}, a12 = {}, a13 = {};

  const size_t rowA0 = (size_t)(mt * 32 + lp);
  const size_t rowA1 = rowA0 + 16;
  {  // phase 1: x @ Wih.T
    const float* X0 = X + rowA0 * Kx + 2 * hi;
    const float* X1 = X + rowA1 * Kx + 2 * hi;
    const float* W0 = Wih + ((size_t)(0 * kH + n0 + lp)) * Kx + 2 * hi;
    const float* W1 = Wih + ((size_t)(1 * kH + n0 + lp)) * Kx + 2 * hi;
    const float* W2 = Wih + ((size_t)(2 * kH + n0 + lp)) * Kx + 2 * hi;
    const float* W3 = Wih + ((size_t)(3 * kH + n0 + lp)) * Kx + 2 * hi;
    gemm_phase(X0, X1, W0, W1, W2, W3, Kx, a00, a01, a02, a03, a10, a11, a12, a13);
  }
  {  // phase 2: h_prev @ Whh.T
    const float* H0 = Hprev + rowA0 * kH + 2 * hi;
    const float* H1 = Hprev + rowA1 * kH + 2 * hi;
    const float* W0 = Whh + ((size_t)(0 * kH + n0 + lp)) * kH + 2 * hi;
    const float* W1 = Whh + ((size_t)(1 * kH + n0 + lp)) * kH + 2 * hi;
    const float* W2 = Whh + ((size_t)(2 * kH + n0 + lp)) * kH + 2 * hi;
    const float* W3 = Whh + ((size_t)(3 * kH + n0 + lp)) * kH + 2 * hi;
    gemm_phase(H0, H1, W0, W1, W2, W3, kH, a00, a01, a02, a03, a10, a11, a12, a13);
  }

  // epilogue: biases + gate nonlinearities + cell/hidden update
  const int n = n0 + lp;
  const float bI = bih[0 * kH + n] + bhh[0 * kH + n];
  const float bF = bih[1 * kH + n] + bhh[1 * kH + n];
  const float bG = bih[2 * kH + n] + bhh[2 * kH + n];
  const float bO = bih[3 * kH + n] + bhh[3 * kH + n];

  #pragma unroll
  for (int r = 0; r < 8; ++r) {
    {  // first 16-row M-tile
      const int m = mt * 32 + r + 8 * hi;
      const size_t idx = (size_t)m * kH + n;
      float iv = sigm(a00[r] + bI);
      float fv = sigm(a01[r] + bF);
      float gv = tanhf(a02[r] + bG);
      float ov = sigm(a03[r] + bO);
      float c2 = fv * Cprev[idx] + iv * gv;
      float h2 = ov * tanhf(c2);
      Cout[idx] = c2;
      Hout[idx] = h2;
    }
    {  // second 16-row M-tile
      const int m = mt * 32 + 16 + r + 8 * hi;
      const size_t idx = (size_t)m * kH + n;
      float iv = sigm(a10[r] + bI);
      float fv = sigm(a11[r] + bF);
      float gv = tanhf(a12[r] + bG);
      float ov = sigm(a13[r] + bO);
      float c2 = fv * Cprev[idx] + iv * gv;
      float h2 = ov * tanhf(c2);
      Cout[idx] = c2;
      Hout[idx] = h2;
    }
  }
}

// ---------------- output head: h3 @ W_out.T + b_out ------------------------
__global__ __launch_bounds__(256) void out_kernel(
    const float* __restrict__ h3, const float* __restrict__ W_out,
    const float* __restrict__ b_out, float* __restrict__ outp) {
  int wid  = blockIdx.x * 8 + (threadIdx.x >> 5);  // 0..511
  int lane = threadIdx.x & 31;
  int b = wid >> 1, o = wid & 1;
  const float* hrow = h3 + (size_t)b * kH;
  const float* wrow = W_out + (size_t)o * kH;
  float acc = 0.f;
  #pragma unroll
  for (int t = 0; t < 8; ++t) {
    int h = (t * 32 + lane) * 4;
    float4 a = *(const float4*)(hrow + h);
    float4 w = *(const float4*)(wrow + h);
    acc += a.x * w.x + a.y * w.y + a.z * w.z + a.w * w.w;
  }
  acc = wave_sum(acc);
  if (lane == 0) outp[b * kOut + o] = acc + b_out[o];
}

// ---------------- host launcher --------------------------------------------
extern "C" void kernel_launch(void* const* d_in, const int* in_sizes, int n_in,
                              void* d_out, int out_size, void* d_ws, size_t ws_size,
                              hipStream_t stream) {
  const float* input = (const float*)d_in[0];
  const float* h0    = (const float*)d_in[1];
  const float* c0    = (const float*)d_in[2];
  const float* all_outputs = (const float*)d_in[3];
  const float* W_emb  = (const float*)d_in[4];
  const float* b_emb  = (const float*)d_in[5];
  const float* W_attn = (const float*)d_in[6];
  const float* b_attn = (const float*)d_in[7];
  const float* Wih0 = (const float*)d_in[8];
  const float* Whh0 = (const float*)d_in[9];
  const float* bih0 = (const float*)d_in[10];
  const float* bhh0 = (const float*)d_in[11];
  const float* Wih_r = (const float*)d_in[12];
  const float* Whh_r = (const float*)d_in[13];
  const float* bih_r = (const float*)d_in[14];
  const float* bhh_r = (const float*)d_in[15];
  const float* W_out = (const float*)d_in[16];
  const float* b_out = (const float*)d_in[17];

  // outputs, concatenated flat in return order
  float* out_final = (float*)d_out;                       // (256, 2)
  float* out_h     = out_final + kB * kOut;               // (4, 256, 1024)
  float* out_c     = out_h + (size_t)kNL * kB * kH;       // (4, 256, 1024)
  float* out_norm  = out_c + (size_t)kNL * kB * kH;       // (256, 256)

  // workspace
  float* xbuf    = (float*)d_ws;                          // (256, 1536)
  float* scoresT = xbuf + (size_t)kB * kKx0;              // (256, 256)
  float* hdot    = scoresT + (size_t)kB * kP;             // (256)

  embed_kernel<<<(kB * kE) / 256, 256, 0, stream>>>(input, W_emb, b_emb, xbuf);
  hdot_kernel<<<kB / 8, 256, 0, stream>>>(h0, W_attn, hdot);
  scores_kernel<<<(kP * kB) / 8, 256, 0, stream>>>(all_outputs, W_attn, b_attn, hdot, scoresT);
  softmax_kernel<<<kB / 8, 256, 0, stream>>>(scoresT, out_norm);
  applied_kernel<<<dim3(kH / 256, kB), 256, 0, stream>>>(all_outputs, out_norm, xbuf);

  // layer 0 (input width 1536), then layers 1..3 (input width 1024)
  // 512 waves (8 M-strips x 64 N-tiles) -> 64 blocks of 8 waves
  lstm_layer_kernel<<<64, 256, 0, stream>>>(xbuf, kKx0, h0, c0,
                                            Wih0, Whh0, bih0, bhh0,
                                            out_h, out_c);
  for (int l = 1; l < kNL; ++l) {
    lstm_layer_kernel<<<64, 256, 0, stream>>>(
        out_h + (size_t)(l - 1) * kB * kH, kH,
        h0 + (size_t)l * kB * kH, c0 + (size_t)l * kB * kH,
        Wih_r + (size_t)(l - 1) * k4H * kH, Whh_r + (size_t)(l - 1) * k4H * kH,
        bih_r + (size_t)(l - 1) * k4H, bhh_r + (size_t)(l - 1) * k4H,
        out_h + (size_t)l * kB * kH, out_c + (size_t)l * kB * kH);
  }
  out_kernel<<<(kB * kOut) / 8, 256, 0, stream>>>(out_h + (size_t)3 * kB * kH,
                                                  W_out, b_out, out_final);
}